// Generator_54451595379094
// MI455X (gfx1250) — compile-verified
//
#include <hip/hip_runtime.h>
#include <hip/hip_bf16.h>
#include <math.h>

// ---------------- model dims ----------------
#define BATCH 256
#define SQ    128
#define FIN   6
#define DE    24     // encoder d_model
#define DH    12     // encoder head dim (2 heads)
#define DD    3      // decoder d_model
#define EMB   10
#define LDR   25     // LDS row stride (odd => conflict-free 16-lane column access)
#define WPAD  32     // padded weight stride (cols 24..31 are zeros)

typedef __attribute__((ext_vector_type(2))) float v2f;
typedef __attribute__((ext_vector_type(8))) float v8f;

// fp32 WMMA: D(16x16) = A(16x4) * B(4x16) + C  -- full fp32 precision on the matrix unit
__device__ __forceinline__ v8f wmma4(v2f a, v2f b, v8f c) {
  return __builtin_amdgcn_wmma_f32_16x16x4_f32(false, a, false, b, (short)0, c,
                                               false, false);
}

// ---------------- parameter pointer table ----------------
struct MhaP { const float *Wq,*bq,*Wk,*bk,*Wv,*bv,*Wo,*bo; };
struct LnP  { const float *g,*b; };
struct EncL { MhaP sa; LnP ln1, ln2; const float *W1,*b1,*W2,*b2; };
struct DecL { MhaP sa, ca; LnP ln1, ln2, ln3; const float *W1,*b1,*W2,*b2; };
struct Net {
  const float *src, *word_emb, *syll_emb;
  EncL enc[4];
  LnP  enc_norm;
  const float *encW, *encb;
  DecL dec[4];
};

// stage a 24x24 weight (+ bias) into LDS as zero-padded 24x32 (+32-entry bias)
__device__ __forceinline__ void stage24(const float* __restrict__ W,
                                        const float* __restrict__ bia,
                                        float* wp, int tid) {
  for (int i = tid; i < DE * WPAD; i += 256) {
    int row = i >> 5, col = i & 31;
    wp[i] = (col < DE) ? W[row * DE + col] : 0.f;
  }
  if (tid < WPAD) wp[DE * WPAD + tid] = (tid < DE) ? bia[tid] : 0.f;
}

// dst(128x24) = src(128x24) @ Wstaged + bias, all in LDS, via fp32 WMMA 16x16x4
__device__ __forceinline__ void proj24(const float* src, float* dst, const float* wp,
                                       int w, int lane) {
  const int half = lane >> 4;
  const int r    = lane & 15;
  for (int t = w; t < 16; t += 8) {
    int tm = (t & 7) * 16, tn = (t >> 3) * 16;
    int col = tn + r;
    v8f acc = {};
    for (int k0 = 0; k0 < DE; k0 += 4) {
      v2f a, bf;
      a.x  = src[(tm + r) * LDR + k0 + (half ? 2 : 0)];
      a.y  = src[(tm + r) * LDR + k0 + (half ? 3 : 1)];
      bf.x = wp[(k0 + (half ? 2 : 0)) * WPAD + col];   // unconditional: padded with zeros
      bf.y = wp[(k0 + (half ? 3 : 1)) * WPAD + col];
      acc = wmma4(a, bf, acc);
    }
    if (col < DE) {
      float bb = wp[DE * WPAD + col];
      for (int j = 0; j < 8; ++j)
        dst[(tm + j + 8 * half) * LDR + col] = acc[j] + bb;
    }
  }
}

// ---------------- kernel 1: embedding + positional encoding ----------------
__global__ __launch_bounds__(256) void embed_kernel(Net net, float* __restrict__ x) {
  int idx = blockIdx.x * 256 + threadIdx.x;          // (b*SQ + s)*DE + c
  if (idx >= BATCH * SQ * DE) return;
  int c  = idx % DE;
  int bs = idx / DE;
  int s  = bs % SQ;
  const float* srow = net.src + (size_t)bs * FIN;
  float v;
  if (c < 4) {
    v = srow[c];
  } else if (c < 4 + EMB) {
    int wi = (int)srow[4];
    v = net.word_emb[(size_t)wi * EMB + (c - 4)];
  } else {
    int si = (int)srow[5];
    v = net.syll_emb[(size_t)si * EMB + (c - 14)];
  }
  // pe[s, 2i] = sin(s*div_i), pe[s, 2i+1] = cos(s*div_i), div_i = exp(-2i*ln(1e4)/DE)
  int i2 = c >> 1;
  float div = __expf(-(float)(2 * i2) * (9.210340371976184f / (float)DE));
  float ang = (float)s * div;
  v += (c & 1) ? cosf(ang) : sinf(ang);
  x[idx] = v;
}

// ---------------- kernel 2: full 4-layer encoder, one block per batch ----------------
__global__ __launch_bounds__(256) void enc_kernel(Net net, float* __restrict__ x,
                                                  float* __restrict__ mem) {
  __shared__ float xs[SQ * LDR];
  __shared__ float qs[SQ * LDR];
  __shared__ float ks[SQ * LDR];
  __shared__ float vs[SQ * LDR + WPAD];  // +pad: unguarded V-fragment reads stay in-bounds
  __shared__ float scr[8 * 16 * 16];     // staged weights / per-wave exp(score) tiles

  const int b    = blockIdx.x;
  const int tid  = threadIdx.x;
  const int w    = tid >> 5;           // wave 0..7
  const int lane = tid & 31;
  const int half = lane >> 4;
  const int r    = lane & 15;
  const int m0   = w * 16;             // this wave's query-row tile

  const float* xg = x + (size_t)b * SQ * DE;
  for (int i = tid; i < SQ * DE; i += 256)
    xs[(i / DE) * LDR + (i % DE)] = xg[i];
  __syncthreads();

  for (int L = 0; L < 4; ++L) {
    const EncL& E = net.enc[L];

    // ---- Q,K,V projections (weights staged into LDS: no divergent global loads)
    stage24(E.sa.Wq, E.sa.bq, scr, tid); __syncthreads();
    proj24(xs, qs, scr, w, lane);        __syncthreads();
    stage24(E.sa.Wk, E.sa.bk, scr, tid); __syncthreads();
    proj24(xs, ks, scr, w, lane);        __syncthreads();
    stage24(E.sa.Wv, E.sa.bv, scr, tid); __syncthreads();
    proj24(xs, vs, scr, w, lane);        __syncthreads();

    // ---- attention: wave w owns query rows [m0, m0+16); two heads sequentially
    const float scale = 0.28867513459481287f;   // 1/sqrt(12)
    for (int h = 0; h < 2; ++h) {
      int hc = h * DH;
      // pass 1: row maxima over all 128 keys (scores via WMMA, max kept in D layout)
      float rmax[8];
      for (int j = 0; j < 8; ++j) rmax[j] = -3.0e38f;
      for (int kt = 0; kt < 8; ++kt) {
        v8f sc = {};
        for (int k0 = 0; k0 < DH; k0 += 4) {
          v2f a, bf;
          a.x  = qs[(m0 + r) * LDR + hc + k0 + (half ? 2 : 0)];
          a.y  = qs[(m0 + r) * LDR + hc + k0 + (half ? 3 : 1)];
          bf.x = ks[(kt * 16 + r) * LDR + hc + k0 + (half ? 2 : 0)];
          bf.y = ks[(kt * 16 + r) * LDR + hc + k0 + (half ? 3 : 1)];
          sc = wmma4(a, bf, sc);
        }
        for (int j = 0; j < 8; ++j) rmax[j] = fmaxf(rmax[j], sc[j] * scale);
      }
      for (int off = 8; off; off >>= 1)
        for (int j = 0; j < 8; ++j)
          rmax[j] = fmaxf(rmax[j], __shfl_xor(rmax[j], off, 32));
      // pass 2: recompute scores, exp, accumulate softmax denominator and O = P@V
      float rsum[8];
      for (int j = 0; j < 8; ++j) rsum[j] = 0.f;
      v8f oacc = {};
      float* myscr = scr + w * 256;
      for (int kt = 0; kt < 8; ++kt) {
        v8f sc = {};
        for (int k0 = 0; k0 < DH; k0 += 4) {
          v2f a, bf;
          a.x  = qs[(m0 + r) * LDR + hc + k0 + (half ? 2 : 0)];
          a.y  = qs[(m0 + r) * LDR + hc + k0 + (half ? 3 : 1)];
          bf.x = ks[(kt * 16 + r) * LDR + hc + k0 + (half ? 2 : 0)];
          bf.y = ks[(kt * 16 + r) * LDR + hc + k0 + (half ? 3 : 1)];
          sc = wmma4(a, bf, sc);
        }
        for (int j = 0; j < 8; ++j) {
          float e = __expf(sc[j] * scale - rmax[j]);
          rsum[j] += e;
          myscr[(j + 8 * half) * 16 + r] = e;   // [m][k] row-major, wave-private
        }
        // same-wave LDS ops are in-order: safe to reload as A fragments.
        // V reads are unguarded: lanes r>=12 feed accumulator columns that are
        // never stored (vs is padded so addresses stay inside the allocation).
        for (int k0 = 0; k0 < 16; k0 += 4) {
          v2f a, bf;
          a.x  = myscr[r * 16 + k0 + (half ? 2 : 0)];
          a.y  = myscr[r * 16 + k0 + (half ? 3 : 1)];
          bf.x = vs[(kt * 16 + k0 + (half ? 2 : 0)) * LDR + hc + r];
          bf.y = vs[(kt * 16 + k0 + (half ? 3 : 1)) * LDR + hc + r];
          oacc = wmma4(a, bf, oacc);
        }
      }
      for (int off = 8; off; off >>= 1)
        for (int j = 0; j < 8; ++j) rsum[j] += __shfl_xor(rsum[j], off, 32);
      // write O back into this head's Q columns (wave-private rows, head-done columns)
      if (r < DH)
        for (int j = 0; j < 8; ++j)
          qs[(m0 + j + 8 * half) * LDR + hc + r] = oacc[j] / rsum[j];
    }
    __syncthreads();

    // ---- output projection: vs = qs @ Wo + bo (staged weights)
    stage24(E.sa.Wo, E.sa.bo, scr, tid); __syncthreads();
    proj24(qs, vs, scr, w, lane);        __syncthreads();

    // ---- residual + LN1 (one thread per sequence row)
    if (tid < SQ) {
      float tmp[DE]; float mean = 0.f;
      for (int c = 0; c < DE; ++c) { tmp[c] = xs[tid*LDR+c] + vs[tid*LDR+c]; mean += tmp[c]; }
      mean *= (1.f / DE); float var = 0.f;
      for (int c = 0; c < DE; ++c) { float d = tmp[c] - mean; var += d * d; }
      var *= (1.f / DE); float rs = __frsqrt_rn(var + 1e-5f);
      for (int c = 0; c < DE; ++c)
        xs[tid*LDR+c] = (tmp[c] - mean) * rs * E.ln1.g[c] + E.ln1.b[c];
    }
    __syncthreads();

    // ---- FFN (24->4->24, VALU: too thin for WMMA) + residual + LN2
    if (tid < SQ) {
      float hb[4];
      for (int f = 0; f < 4; ++f) {
        float s = E.b1[f];
        for (int c = 0; c < DE; ++c) s += xs[tid*LDR+c] * E.W1[c*4+f];
        hb[f] = fmaxf(s, 0.f);
      }
      float tmp[DE]; float mean = 0.f;
      for (int c = 0; c < DE; ++c) {
        float s = E.b2[c];
        for (int f = 0; f < 4; ++f) s += hb[f] * E.W2[f*DE+c];
        tmp[c] = xs[tid*LDR+c] + s; mean += tmp[c];
      }
      mean *= (1.f / DE); float var = 0.f;
      for (int c = 0; c < DE; ++c) { float d = tmp[c] - mean; var += d * d; }
      var *= (1.f / DE); float rs = __frsqrt_rn(var + 1e-5f);
      for (int c = 0; c < DE; ++c)
        xs[tid*LDR+c] = (tmp[c] - mean) * rs * E.ln2.g[c] + E.ln2.b[c];
    }
    __syncthreads();
  }

  // ---- final encoder norm + projection to memory [SQ, 3]
  if (tid < SQ) {
    float tmp[DE]; float mean = 0.f;
    for (int c = 0; c < DE; ++c) { tmp[c] = xs[tid*LDR+c]; mean += tmp[c]; }
    mean *= (1.f / DE); float var = 0.f;
    for (int c = 0; c < DE; ++c) { float d = tmp[c] - mean; var += d * d; }
    var *= (1.f / DE); float rs = __frsqrt_rn(var + 1e-5f);
    float nrm[DE];
    for (int c = 0; c < DE; ++c)
      nrm[c] = (tmp[c] - mean) * rs * net.enc_norm.g[c] + net.enc_norm.b[c];
    float* mrow = mem + ((size_t)b * SQ + tid) * DD;
    for (int c = 0; c < DD; ++c) {
      float s = net.encb[c];
      for (int k = 0; k < DE; ++k) s += nrm[k] * net.encW[k*DD+c];
      mrow[c] = s;
    }
  }
}

// ---------------- LN over 3 elements (gamma/beta from LDS) ----------------
__device__ __forceinline__ void ln3(float v0, float v1, float v2,
                                    const float* g, const float* bb,
                                    float& o0, float& o1, float& o2) {
  float m  = (v0 + v1 + v2) * (1.f / 3.f);
  float d0 = v0 - m, d1 = v1 - m, d2 = v2 - m;
  float rs = __frsqrt_rn((d0*d0 + d1*d1 + d2*d2) * (1.f / 3.f) + 1e-5f);
  o0 = d0 * rs * g[0] + bb[0];
  o1 = d1 * rs * g[1] + bb[1];
  o2 = d2 * rs * g[2] + bb[2];
}

// decoder LDS weight-cache layout (per layer, stride DWL):
//  0 saWq9  9 saWk9 18 saWv9 27 saWo9 36 sabq3 39 sabk3 42 sabv3 45 sabo3
// 48 caWq9 57 caWk9 66 caWv9 75 caWo9 84 cabq3 87 cabk3 90 cabv3 93 cabo3
// 96 ln1g 99 ln1b 102 ln2g 105 ln2b 108 ln3g 111 ln3b
// 114 W1(12) 126 b1(4) 130 W2(12) 142 b2(3)
#define DWL 160

// ---------------- kernel 3: KV-cached autoregressive decoder ----------------
// one block = one batch element = ONE wave32 => shuffle-only reductions; all
// weights cached in LDS so the 127-step serial chain is pure VALU + DS.
__global__ __launch_bounds__(32) void dec_kernel(Net net, const float* __restrict__ mem,
                                                 float* __restrict__ out) {
  __shared__ float ms[SQ * DD];
  __shared__ float ck[4][SQ * DD];   // cross-attn K of memory (precomputed per layer)
  __shared__ float cv[4][SQ * DD];
  __shared__ float sk[4][SQ * DD];   // self-attn KV cache
  __shared__ float sv[4][SQ * DD];
  __shared__ float dw[4 * DWL];      // all decoder weights, LDS-resident

  const int b = blockIdx.x;
  const int lane = threadIdx.x;

  // cache all decoder weights in LDS (once per block)
  auto cp = [&](int off, const float* p, int n) {
    for (int t = lane; t < n; t += 32) dw[off + t] = p[t];
  };
  for (int l = 0; l < 4; ++l) {
    const DecL& D = net.dec[l]; int o = l * DWL;
    cp(o+0,  D.sa.Wq, 9); cp(o+9,  D.sa.Wk, 9); cp(o+18, D.sa.Wv, 9); cp(o+27, D.sa.Wo, 9);
    cp(o+36, D.sa.bq, 3); cp(o+39, D.sa.bk, 3); cp(o+42, D.sa.bv, 3); cp(o+45, D.sa.bo, 3);
    cp(o+48, D.ca.Wq, 9); cp(o+57, D.ca.Wk, 9); cp(o+66, D.ca.Wv, 9); cp(o+75, D.ca.Wo, 9);
    cp(o+84, D.ca.bq, 3); cp(o+87, D.ca.bk, 3); cp(o+90, D.ca.bv, 3); cp(o+93, D.ca.bo, 3);
    cp(o+96, D.ln1.g, 3); cp(o+99, D.ln1.b, 3); cp(o+102, D.ln2.g, 3); cp(o+105, D.ln2.b, 3);
    cp(o+108, D.ln3.g, 3); cp(o+111, D.ln3.b, 3);
    cp(o+114, D.W1, 12); cp(o+126, D.b1, 4); cp(o+130, D.W2, 12); cp(o+142, D.b2, 3);
  }
  const float* mrow = mem + (size_t)b * SQ * DD;
  for (int i = lane; i < SQ * DD; i += 32) ms[i] = mrow[i];
  __syncthreads();

  // precompute cross-attention K/V (memory is fixed for all 127 steps)
  for (int l = 0; l < 4; ++l) {
    int o = l * DWL;
    for (int j = lane; j < SQ; j += 32) {
      float m0v = ms[j*DD+0], m1v = ms[j*DD+1], m2v = ms[j*DD+2];
      for (int c = 0; c < DD; ++c) {
        ck[l][j*DD+c] = dw[o+87+c] + m0v*dw[o+57+c] + m1v*dw[o+60+c] + m2v*dw[o+63+c];
        cv[l][j*DD+c] = dw[o+90+c] + m0v*dw[o+66+c] + m1v*dw[o+69+c] + m2v*dw[o+72+c];
      }
    }
  }
  __syncthreads();

  // passthrough columns + tgt[:,0] = 0
  float* ob = out + (size_t)b * SQ * 5;
  const float* sb = net.src + (size_t)b * SQ * FIN;
  for (int s = lane; s < SQ; s += 32) {
    ob[s*5+3] = sb[s*FIN+4];
    ob[s*5+4] = sb[s*FIN+5];
  }
  if (lane < DD) ob[lane] = 0.f;

  const float iscale = 0.5773502691896258f;  // 1/sqrt(3)
  float t0 = 0.f, t1 = 0.f, t2 = 0.f;        // buf[0] = 0

  for (int i = 0; i < SQ - 1; ++i) {
    float x0 = t0, x1 = t1, x2 = t2;
    for (int l = 0; l < 4; ++l) {
      const int o = l * DWL;
      // ---- self-attention (redundant scalar math on all lanes; no divergence)
      float q0 = dw[o+36] + x0*dw[o+0] + x1*dw[o+3] + x2*dw[o+6];
      float q1 = dw[o+37] + x0*dw[o+1] + x1*dw[o+4] + x2*dw[o+7];
      float q2 = dw[o+38] + x0*dw[o+2] + x1*dw[o+5] + x2*dw[o+8];
      if (lane == 0) {
        for (int c = 0; c < DD; ++c) {
          sk[l][i*DD+c] = dw[o+39+c] + x0*dw[o+9+c]  + x1*dw[o+12+c] + x2*dw[o+15+c];
          sv[l][i*DD+c] = dw[o+42+c] + x0*dw[o+18+c] + x1*dw[o+21+c] + x2*dw[o+24+c];
        }
      }
      __syncthreads();   // single-wave WG: barrier op is S_NOP, just orders LDS
      float s4[4], mloc = -3.0e38f;
      for (int u = 0; u < 4; ++u) {
        int j = lane + u * 32;
        float s = -3.0e38f;
        if (j <= i)
          s = (q0*sk[l][j*DD] + q1*sk[l][j*DD+1] + q2*sk[l][j*DD+2]) * iscale;
        s4[u] = s; mloc = fmaxf(mloc, s);
      }
      for (int off = 16; off; off >>= 1) mloc = fmaxf(mloc, __shfl_xor(mloc, off, 32));
      float ps = 0.f, o0 = 0.f, o1 = 0.f, o2 = 0.f;
      for (int u = 0; u < 4; ++u) {
        int j = lane + u * 32;
        if (j <= i) {
          float e = __expf(s4[u] - mloc);
          ps += e; o0 += e*sv[l][j*DD]; o1 += e*sv[l][j*DD+1]; o2 += e*sv[l][j*DD+2];
        }
      }
      for (int off = 16; off; off >>= 1) {
        ps += __shfl_xor(ps, off, 32);
        o0 += __shfl_xor(o0, off, 32);
        o1 += __shfl_xor(o1, off, 32);
        o2 += __shfl_xor(o2, off, 32);
      }
      float inv = 1.f / ps; o0 *= inv; o1 *= inv; o2 *= inv;
      float a0 = dw[o+45] + o0*dw[o+27] + o1*dw[o+30] + o2*dw[o+33];
      float a1 = dw[o+46] + o0*dw[o+28] + o1*dw[o+31] + o2*dw[o+34];
      float a2 = dw[o+47] + o0*dw[o+29] + o1*dw[o+32] + o2*dw[o+35];
      ln3(x0 + a0, x1 + a1, x2 + a2, dw+o+96, dw+o+99, x0, x1, x2);

      // ---- cross-attention over all 128 memory positions
      q0 = dw[o+84] + x0*dw[o+48] + x1*dw[o+51] + x2*dw[o+54];
      q1 = dw[o+85] + x0*dw[o+49] + x1*dw[o+52] + x2*dw[o+55];
      q2 = dw[o+86] + x0*dw[o+50] + x1*dw[o+53] + x2*dw[o+56];
      mloc = -3.0e38f;
      for (int u = 0; u < 4; ++u) {
        int j = lane + u * 32;
        float s = (q0*ck[l][j*DD] + q1*ck[l][j*DD+1] + q2*ck[l][j*DD+2]) * iscale;
        s4[u] = s; mloc = fmaxf(mloc, s);
      }
      for (int off = 16; off; off >>= 1) mloc = fmaxf(mloc, __shfl_xor(mloc, off, 32));
      ps = 0.f; o0 = 0.f; o1 = 0.f; o2 = 0.f;
      for (int u = 0; u < 4; ++u) {
        int j = lane + u * 32;
        float e = __expf(s4[u] - mloc);
        ps += e; o0 += e*cv[l][j*DD]; o1 += e*cv[l][j*DD+1]; o2 += e*cv[l][j*DD+2];
      }
      for (int off = 16; off; off >>= 1) {
        ps += __shfl_xor(ps, off, 32);
        o0 += __shfl_xor(o0, off, 32);
        o1 += __shfl_xor(o1, off, 32);
        o2 += __shfl_xor(o2, off, 32);
      }
      inv = 1.f / ps; o0 *= inv; o1 *= inv; o2 *= inv;
      a0 = dw[o+93] + o0*dw[o+75] + o1*dw[o+78] + o2*dw[o+81];
      a1 = dw[o+94] + o0*dw[o+76] + o1*dw[o+79] + o2*dw[o+82];
      a2 = dw[o+95] + o0*dw[o+77] + o1*dw[o+80] + o2*dw[o+83];
      ln3(x0 + a0, x1 + a1, x2 + a2, dw+o+102, dw+o+105, x0, x1, x2);

      // ---- FFN 3->4->3
      float hb[4];
      for (int f = 0; f < 4; ++f)
        hb[f] = fmaxf(dw[o+126+f] + x0*dw[o+114+f] + x1*dw[o+118+f] + x2*dw[o+122+f], 0.f);
      float y0 = dw[o+142], y1 = dw[o+143], y2 = dw[o+144];
      for (int f = 0; f < 4; ++f) {
        y0 += hb[f]*dw[o+130+f*3+0]; y1 += hb[f]*dw[o+130+f*3+1]; y2 += hb[f]*dw[o+130+f*3+2];
      }
      ln3(x0 + y0, x1 + y1, x2 + y2, dw+o+108, dw+o+111, x0, x1, x2);
    }
    t0 = x0; t1 = x1; t2 = x2;                 // buf[i+1] = out[i]
    if (lane == 0) {
      ob[(i+1)*5+0] = t0; ob[(i+1)*5+1] = t1; ob[(i+1)*5+2] = t2;
    }
  }
}

// ---------------- host ----------------
extern "C" void kernel_launch(void* const* d_in, const int* in_sizes, int n_in,
                              void* d_out, int out_size, void* d_ws, size_t ws_size,
                              hipStream_t stream) {
  (void)n_in; (void)out_size; (void)ws_size;
  Net net;
  int k = 0;
  auto nx = [&]() { return (const float*)d_in[k++]; };

  // Detect flattening convention: jax-pytree (sorted keys) puts dec W1 (12 elems)
  // first; dict-insertion order puts src (256*128*6 elems) first.
  bool sorted_order = (in_sizes[0] != BATCH * SQ * FIN);
  if (sorted_order) {
    for (int l = 0; l < 4; ++l) {                       // dec_layers, sorted keys
      DecL& D = net.dec[l];
      D.W1 = nx(); D.W2 = nx(); D.b1 = nx(); D.b2 = nx();
      D.ca.Wk = nx(); D.ca.Wo = nx(); D.ca.Wq = nx(); D.ca.Wv = nx();
      D.ca.bk = nx(); D.ca.bo = nx(); D.ca.bq = nx(); D.ca.bv = nx();
      D.ln1.b = nx(); D.ln1.g = nx(); D.ln2.b = nx(); D.ln2.g = nx();
      D.ln3.b = nx(); D.ln3.g = nx();
      D.sa.Wk = nx(); D.sa.Wo = nx(); D.sa.Wq = nx(); D.sa.Wv = nx();
      D.sa.bk = nx(); D.sa.bo = nx(); D.sa.bq = nx(); D.sa.bv = nx();
    }
    for (int l = 0; l < 4; ++l) {                       // enc_layers, sorted keys
      EncL& E = net.enc[l];
      E.W1 = nx(); E.W2 = nx(); E.b1 = nx(); E.b2 = nx();
      E.ln1.b = nx(); E.ln1.g = nx(); E.ln2.b = nx(); E.ln2.g = nx();
      E.sa.Wk = nx(); E.sa.Wo = nx(); E.sa.Wq = nx(); E.sa.Wv = nx();
      E.sa.bk = nx(); E.sa.bo = nx(); E.sa.bq = nx(); E.sa.bv = nx();
    }
    net.enc_norm.b = nx(); net.enc_norm.g = nx();
    net.encW = nx(); net.encb = nx();
    net.syll_emb = nx(); net.word_emb = nx();
    net.src = nx();
  } else {                                              // dict insertion order
    net.src = nx(); net.word_emb = nx(); net.syll_emb = nx();
    for (int l = 0; l < 4; ++l) {
      EncL& E = net.enc[l];
      E.sa.Wq = nx(); E.sa.bq = nx(); E.sa.Wk = nx(); E.sa.bk = nx();
      E.sa.Wv = nx(); E.sa.bv = nx(); E.sa.Wo = nx(); E.sa.bo = nx();
      E.ln1.g = nx(); E.ln1.b = nx(); E.ln2.g = nx(); E.ln2.b = nx();
      E.W1 = nx(); E.b1 = nx(); E.W2 = nx(); E.b2 = nx();
    }
    net.enc_norm.g = nx(); net.enc_norm.b = nx();
    net.encW = nx(); net.encb = nx();
    for (int l = 0; l < 4; ++l) {
      DecL& D = net.dec[l];
      D.sa.Wq = nx(); D.sa.bq = nx(); D.sa.Wk = nx(); D.sa.bk = nx();
      D.sa.Wv = nx(); D.sa.bv = nx(); D.sa.Wo = nx(); D.sa.bo = nx();
      D.ca.Wq = nx(); D.ca.bq = nx(); D.ca.Wk = nx(); D.ca.bk = nx();
      D.ca.Wv = nx(); D.ca.bv = nx(); D.ca.Wo = nx(); D.ca.bo = nx();
      D.ln1.g = nx(); D.ln1.b = nx(); D.ln2.g = nx(); D.ln2.b = nx();
      D.ln3.g = nx(); D.ln3.b = nx();
      D.W1 = nx(); D.b1 = nx(); D.W2 = nx(); D.b2 = nx();
    }
  }

  float* x   = (float*)d_ws;                            // [B,S,24]
  float* mem = x + (size_t)BATCH * SQ * DE;             // [B,S,3]

  embed_kernel<<<(BATCH * SQ * DE + 255) / 256, 256, 0, stream>>>(net, x);
  enc_kernel<<<BATCH, 256, 0, stream>>>(net, x, mem);
  dec_kernel<<<BATCH, 32, 0, stream>>>(net, mem, (float*)d_out);
}